// SequenceMemoryUpdater_54107997995380
// MI455X (gfx1250) — compile-verified
//
#include <hip/hip_runtime.h>
#include <cstddef>

// ---------------------------------------------------------------------------
// TGN memory updater for MI455X (gfx1250, wave32, WMMA f32 16x16x4)
//
//   h        = memory_table[ids]                  (gather,  [B,256])
//   gi       = msgs @ W_ih^T + b_ih               ([B,768])
//   gh       = h    @ W_hh^T + b_hh               ([B,768])
//   r,z      = sigmoid(gi_r+gh_r), sigmoid(gi_z+gh_z)
//   n        = tanh(gi_n + r*gh_n)
//   new_h    = (1-z)*n + z*h
//   out_mem  = memory_table with rows[ids] = new_h   (clone via D2D memcpy)
//   out_lu   = last_update  with [ids] = timestamps
// ---------------------------------------------------------------------------

#define N_NODES 500000
#define MEM_DIM 256
#define MSG_DIM 512

// LDS row strides: stride % 64 == 4 floats -> the 16 A-matrix lanes map to 16
// distinct bank-quads (conflict-free ds_load_b64), and stride*4 % 16 == 0 so
// float4 staging stores stay aligned.
#define XS_STRIDE (MSG_DIM + 4)   // 516 floats
#define HS_STRIDE (MEM_DIM + 4)   // 260 floats

typedef __attribute__((ext_vector_type(2))) float v2f;
typedef __attribute__((ext_vector_type(8))) float v8f;

__device__ __forceinline__ float sigmoid_f(float x) {
    return 1.0f / (1.0f + __expf(-x));
}
// tanh(x) = 2*sigmoid(2x) - 1 ; saturates correctly at +-1 for large |x|
__device__ __forceinline__ float tanh_f(float x) {
    return 2.0f / (1.0f + __expf(-2.0f * x)) - 1.0f;
}

__device__ __forceinline__ v8f wmma_f32(v2f a, v2f b, v8f c) {
    // D = A(16x4) * B(4x16) + C(16x16), full fp32
    return __builtin_amdgcn_wmma_f32_16x16x4_f32(
        /*neg_a=*/false, a, /*neg_b=*/false, b,
        /*c_mod=*/(short)0, c, /*reuse_a=*/false, /*reuse_b=*/false);
}

// One workgroup = one tile of 16 batch rows. 256 threads = 8 waves.
// Wave w computes column tiles t = 2w and 2w+1 (t in 0..15 covers d=0..255),
// and for each t accumulates all six gate fragments so the GRU gate math is
// purely elementwise on matching fragment layouts.
__global__ void __launch_bounds__(256)
tgn_gru_update_kernel(float* __restrict__ out_mem,          // [N_NODES, 256]
                      float* __restrict__ out_lu,           // [N_NODES]
                      const int* __restrict__ node_ids,     // [B]
                      const float* __restrict__ messages,   // [B, 512]
                      const float* __restrict__ timestamps, // [B]
                      const float* __restrict__ W_ih,       // [768, 512]
                      const float* __restrict__ W_hh,       // [768, 256]
                      const float* __restrict__ b_ih,       // [768]
                      const float* __restrict__ b_hh,       // [768]
                      const float* __restrict__ memory_table, // [N_NODES,256]
                      int B) {
    __shared__ __align__(16) float Xs[16 * XS_STRIDE];
    __shared__ __align__(16) float Hs[16 * HS_STRIDE];
    __shared__ int s_nid[16];

    const int tid  = threadIdx.x;
    const int tile = blockIdx.x;
    const int row0 = tile * 16;

    // ---- stage node ids; scatter timestamps (ids are unique -> race-free)
    if (tid < 16) {
        const int g = row0 + tid;
        int nid = -1;
        if (g < B) {
            nid = node_ids[g];
            out_lu[nid] = timestamps[g];
        }
        s_nid[tid] = nid;
    }
    __syncthreads();

    // ---- cooperatively stage X tile (16x512) into LDS, float4-coalesced
    for (int i = tid; i < 16 * (MSG_DIM / 4); i += 256) {
        const int r  = i >> 7;          // /128
        const int c4 = i & 127;
        float4 v = make_float4(0.f, 0.f, 0.f, 0.f);
        if (row0 + r < B)
            v = *(const float4*)(messages + (size_t)(row0 + r) * MSG_DIM + c4 * 4);
        *(float4*)(&Xs[r * XS_STRIDE + c4 * 4]) = v;
    }
    // ---- stage gathered H tile (16x256) into LDS
    for (int i = tid; i < 16 * (MEM_DIM / 4); i += 256) {
        const int r  = i >> 6;          // /64
        const int c4 = i & 63;
        const int nid = s_nid[r];
        float4 v = make_float4(0.f, 0.f, 0.f, 0.f);
        if (nid >= 0)
            v = *(const float4*)(memory_table + (size_t)nid * MEM_DIM + c4 * 4);
        *(float4*)(&Hs[r * HS_STRIDE + c4 * 4]) = v;
    }
    __syncthreads();

    const int lane = tid & 31;
    const int l16  = lane & 15;         // N / M column within fragment
    const int hl   = lane >> 4;         // lane half -> K sub-offset 2*hl
    const int wv   = tid >> 5;          // wave id 0..7

    for (int j = 0; j < 2; ++j) {
        const int t  = 2 * wv + j;      // column tile 0..15
        const int nr = 16 * t;          // d-range base within [0,256)

        v8f ir = {}, iz = {}, in_ = {}, hr = {}, hz = {}, hn = {};

        // ---- GI = X @ W_ih^T over K=512 (A from LDS, B rows from L2)
        {
            const float* Wr = W_ih + (size_t)(nr +       l16) * MSG_DIM + 2 * hl;
            const float* Wz = W_ih + (size_t)(nr + 256 + l16) * MSG_DIM + 2 * hl;
            const float* Wn = W_ih + (size_t)(nr + 512 + l16) * MSG_DIM + 2 * hl;
            const float* Xrow = &Xs[l16 * XS_STRIDE + 2 * hl];
#pragma unroll 4
            for (int k0 = 0; k0 < MSG_DIM; k0 += 4) {
                const v2f a  = *(const v2f*)(Xrow + k0);
                const v2f br = *(const v2f*)(Wr + k0);
                const v2f bz = *(const v2f*)(Wz + k0);
                const v2f bn = *(const v2f*)(Wn + k0);
                ir  = wmma_f32(a, br, ir);
                iz  = wmma_f32(a, bz, iz);
                in_ = wmma_f32(a, bn, in_);
            }
        }
        // ---- GH = H @ W_hh^T over K=256
        {
            const float* Wr = W_hh + (size_t)(nr +       l16) * MEM_DIM + 2 * hl;
            const float* Wz = W_hh + (size_t)(nr + 256 + l16) * MEM_DIM + 2 * hl;
            const float* Wn = W_hh + (size_t)(nr + 512 + l16) * MEM_DIM + 2 * hl;
            const float* Hrow = &Hs[l16 * HS_STRIDE + 2 * hl];
#pragma unroll 4
            for (int k0 = 0; k0 < MEM_DIM; k0 += 4) {
                const v2f a  = *(const v2f*)(Hrow + k0);
                const v2f br = *(const v2f*)(Wr + k0);
                const v2f bz = *(const v2f*)(Wz + k0);
                const v2f bn = *(const v2f*)(Wn + k0);
                hr = wmma_f32(a, br, hr);
                hz = wmma_f32(a, bz, hz);
                hn = wmma_f32(a, bn, hn);
            }
        }

        // ---- GRU gates: fragment layouts of all six accumulators match
        // (element e, lane) -> (m = e + 8*hl, d = nr + l16)
        const float bir = b_ih[nr +       l16];
        const float biz = b_ih[nr + 256 + l16];
        const float bin = b_ih[nr + 512 + l16];
        const float bhr = b_hh[nr +       l16];
        const float bhz = b_hh[nr + 256 + l16];
        const float bhn = b_hh[nr + 512 + l16];

#pragma unroll
        for (int e = 0; e < 8; ++e) {
            const int m   = e + 8 * hl;
            const int row = s_nid[m];
            const float rr = sigmoid_f((ir[e] + bir) + (hr[e] + bhr));
            const float zz = sigmoid_f((iz[e] + biz) + (hz[e] + bhz));
            const float nv = tanh_f((in_[e] + bin) + rr * (hn[e] + bhn));
            const float hv = Hs[m * HS_STRIDE + nr + l16];
            const float nh = (1.0f - zz) * nv + zz * hv;
            if (row >= 0)
                out_mem[(size_t)row * MEM_DIM + nr + l16] = nh;
        }
    }
}

extern "C" void kernel_launch(void* const* d_in, const int* in_sizes, int n_in,
                              void* d_out, int out_size, void* d_ws, size_t ws_size,
                              hipStream_t stream) {
    const float* memory_table = (const float*)d_in[0];
    const float* last_update  = (const float*)d_in[1];
    const int*   node_ids     = (const int*)d_in[2];
    const float* messages     = (const float*)d_in[3];
    const float* timestamps   = (const float*)d_in[4];
    const float* W_ih         = (const float*)d_in[5];
    const float* W_hh         = (const float*)d_in[6];
    const float* b_ih         = (const float*)d_in[7];
    const float* b_hh         = (const float*)d_in[8];

    const int B = in_sizes[2];

    float* out_mem = (float*)d_out;
    float* out_lu  = out_mem + (size_t)N_NODES * MEM_DIM;

    // Clone memory_table / last_update into the outputs (stream-ordered, so
    // the GRU kernel's scatter lands after the copy).
    hipMemcpyAsync(out_mem, memory_table,
                   (size_t)N_NODES * MEM_DIM * sizeof(float),
                   hipMemcpyDeviceToDevice, stream);
    hipMemcpyAsync(out_lu, last_update,
                   (size_t)N_NODES * sizeof(float),
                   hipMemcpyDeviceToDevice, stream);

    const int tiles = (B + 15) / 16;   // 12500 workgroups for B=200000
    tgn_gru_update_kernel<<<tiles, 256, 0, stream>>>(
        out_mem, out_lu, node_ids, messages, timestamps,
        W_ih, W_hh, b_ih, b_hh, memory_table, B);
}